// GatEncoder_78795470012820
// MI455X (gfx1250) — compile-verified
//
#include <hip/hip_runtime.h>
#include <hip/hip_bf16.h>

typedef __attribute__((ext_vector_type(2))) float v2f;
typedef __attribute__((ext_vector_type(8))) float v8f;

#define HID_   128
#define HEADS  4
#define GRAPHS 256
#define NHID_  256
#define NOUT_  256
#define NEG_SLOPE 0.2f

// ---- monotone float <-> uint encoding for atomicMax on floats ----
__device__ __forceinline__ unsigned enc_f32(float f) {
  unsigned u = __float_as_uint(f);
  return (u & 0x80000000u) ? ~u : (u | 0x80000000u);
}
__device__ __forceinline__ float dec_f32(unsigned u) {
  return __uint_as_float((u & 0x80000000u) ? (u & 0x7FFFFFFFu) : ~u);
}
// enc(-FLT_MAX) = ~0xFF7FFFFF = 0x00800000 ; anything real beats it.
#define ENC_NEG_MAX 0x00800000u

// ===================================================================
// Generic fp32 GEMM via V_WMMA_F32_16X16X4_F32: out[M,N] = A[M,K] @ W[K,N] (+bias)(+relu)
// One wave32 computes one 16x16 output tile.  M,N multiples of 16; K multiple of 4.
// ===================================================================
__global__ void k_gemm16(const float* __restrict__ A, const float* __restrict__ W,
                         float* __restrict__ out, int M, int N, int K,
                         const float* __restrict__ bias, int relu)
{
  long long gt = (long long)blockIdx.x * blockDim.x + threadIdx.x;
  int wave = (int)(gt >> 5);
  int ntiles = N >> 4;
  int mt = wave / ntiles;
  int nt = wave - mt * ntiles;
  if (mt * 16 >= M) return;            // wave-uniform: EXEC stays all-ones for WMMA

  int lane = threadIdx.x & 31;
  int half = lane >> 4;                // 0: lanes 0-15, 1: lanes 16-31
  int l    = lane & 15;

  const float* Ar = A + (size_t)(mt * 16 + l) * K;   // A row for this lane (M=l)
  const float* Wc = W + (size_t)(nt * 16 + l);       // B column for this lane (N=l)

  v8f acc = {};
  for (int k = 0; k < K; k += 4) {
    v2f a, b;
    // A 16x4 layout: VGPR0 = K=k+2*half, VGPR1 = K=k+2*half+1 (ISA 7.12.2)
    a.x = Ar[k + 2 * half];
    a.y = Ar[k + 2 * half + 1];
    // B 4x16 layout mirrors: rows striped across lanes within a VGPR
    b.x = Wc[(size_t)(k + 2 * half) * N];
    b.y = Wc[(size_t)(k + 2 * half + 1) * N];
    acc = __builtin_amdgcn_wmma_f32_16x16x4_f32(false, a, false, b, (short)0, acc,
                                                false, false);
  }

  float bv = bias ? bias[nt * 16 + l] : 0.0f;
  // C/D layout: VGPR v -> M = v + 8*half, N = l
  float* orow = out + (size_t)(mt * 16 + half * 8) * N + nt * 16 + l;
  #pragma unroll
  for (int v = 0; v < 8; ++v) {
    float val = acc[v] + bv;
    if (relu) val = fmaxf(val, 0.0f);
    orow[(size_t)v * N] = val;
  }
}

// ===================================================================
// Edge score: one wave per edge.  Lane c covers 4 channels (float4);
// lanes [8h, 8h+8) cover head h.  Reduce per head, atomicMax per (dst,head).
// ===================================================================
__global__ void k_edge_scores(const float* __restrict__ hs, const float* __restrict__ hd,
                              const int* __restrict__ ei, int n_edges, int n_nodes,
                              const float* __restrict__ att,
                              float* __restrict__ e_out, unsigned* __restrict__ maxenc)
{
  long long gt = (long long)blockIdx.x * blockDim.x + threadIdx.x;
  int e = (int)(gt >> 5);
  int total = n_edges + n_nodes;
  if (e >= total) return;
  int lane = threadIdx.x & 31;

  int src, dst;
  if (e < n_edges) { src = ei[e]; dst = ei[n_edges + e]; }
  else             { src = e - n_edges; dst = src; }     // self-loop

  float4 s4 = ((const float4*)(hs + (size_t)src * HID_))[lane];
  float4 d4 = ((const float4*)(hd + (size_t)dst * HID_))[lane];
  float4 a4 = ((const float4*)att)[lane];

  float t0 = s4.x + d4.x; t0 = t0 > 0.0f ? t0 : NEG_SLOPE * t0;
  float t1 = s4.y + d4.y; t1 = t1 > 0.0f ? t1 : NEG_SLOPE * t1;
  float t2 = s4.z + d4.z; t2 = t2 > 0.0f ? t2 : NEG_SLOPE * t2;
  float t3 = s4.w + d4.w; t3 = t3 > 0.0f ? t3 : NEG_SLOPE * t3;
  float p = t0 * a4.x + t1 * a4.y + t2 * a4.z + t3 * a4.w;

  // reduce within 8-lane head group
  p += __shfl_xor(p, 1, 32);
  p += __shfl_xor(p, 2, 32);
  p += __shfl_xor(p, 4, 32);

  if ((lane & 7) == 0) {
    int h = lane >> 3;
    e_out[(size_t)e * HEADS + h] = p;
    atomicMax(&maxenc[(size_t)dst * HEADS + h], enc_f32(p));
  }
}

// ===================================================================
// exp(e - max[dst]) in place, accumulate denom per (dst,head)
// ===================================================================
__global__ void k_edge_exp(float* __restrict__ ebuf, const unsigned* __restrict__ maxenc,
                           const int* __restrict__ ei, int n_edges, int n_nodes,
                           float* __restrict__ denom)
{
  long long idx = (long long)blockIdx.x * blockDim.x + threadIdx.x;
  long long total = (long long)(n_edges + n_nodes) * HEADS;
  if (idx >= total) return;
  int e = (int)(idx >> 2), h = (int)(idx & 3);
  int dst = (e < n_edges) ? ei[n_edges + e] : (e - n_edges);
  float m = dec_f32(maxenc[(size_t)dst * HEADS + h]);
  float ex = expf(ebuf[idx] - m);
  ebuf[idx] = ex;
  atomicAdd(&denom[(size_t)dst * HEADS + h], ex);
}

// ===================================================================
// Aggregation: one wave per edge; lane c adds alpha*hs[src] (4 ch) into agg[dst]
// ===================================================================
__global__ void k_edge_aggr(const float* __restrict__ hs, const float* __restrict__ ebuf,
                            const float* __restrict__ denom, const int* __restrict__ ei,
                            int n_edges, int n_nodes, float* __restrict__ agg)
{
  long long gt = (long long)blockIdx.x * blockDim.x + threadIdx.x;
  int e = (int)(gt >> 5);
  int total = n_edges + n_nodes;
  if (e >= total) return;
  int lane = threadIdx.x & 31;

  int src, dst;
  if (e < n_edges) { src = ei[e]; dst = ei[n_edges + e]; }
  else             { src = e - n_edges; dst = src; }

  int h = lane >> 3;
  float den   = denom[(size_t)dst * HEADS + h];
  float alpha = ebuf[(size_t)e * HEADS + h] / fmaxf(den, 1e-16f);

  float4 s4 = ((const float4*)(hs + (size_t)src * HID_))[lane];
  float* o = agg + (size_t)dst * HID_ + lane * 4;
  atomicAdd(o + 0, s4.x * alpha);
  atomicAdd(o + 1, s4.y * alpha);
  atomicAdd(o + 2, s4.z * alpha);
  atomicAdd(o + 3, s4.w * alpha);
}

// ===================================================================
// out += bias, optional relu (in place)
// ===================================================================
__global__ void k_bias_act(float* __restrict__ h, const float* __restrict__ b,
                           int n_nodes, int relu)
{
  long long idx = (long long)blockIdx.x * blockDim.x + threadIdx.x;
  long long total = (long long)n_nodes * HID_;
  if (idx >= total) return;
  float v = h[idx] + b[idx & (HID_ - 1)];
  if (relu) v = fmaxf(v, 0.0f);
  h[idx] = v;
}

__global__ void k_fill_u32(unsigned* __restrict__ p, unsigned val, long long n)
{
  long long i = (long long)blockIdx.x * blockDim.x + threadIdx.x;
  if (i < n) p[i] = val;
}

// ===================================================================
// global mean pool (sums + counts), then divide
// ===================================================================
__global__ void k_pool(const float* __restrict__ h, const int* __restrict__ batch,
                       float* __restrict__ sums, float* __restrict__ cnts, int n_nodes)
{
  long long idx = (long long)blockIdx.x * blockDim.x + threadIdx.x;
  long long total = (long long)n_nodes * HID_;
  if (idx >= total) return;
  int node = (int)(idx >> 7);
  int c    = (int)(idx & (HID_ - 1));
  int g = batch[node];
  atomicAdd(&sums[(size_t)g * HID_ + c], h[idx]);
  if (c == 0) atomicAdd(&cnts[g], 1.0f);
}

__global__ void k_pool_div(const float* __restrict__ sums, const float* __restrict__ cnts,
                           float* __restrict__ g)
{
  int idx = blockIdx.x * blockDim.x + threadIdx.x;
  if (idx >= GRAPHS * HID_) return;
  g[idx] = sums[idx] / fmaxf(cnts[idx >> 7], 1.0f);
}

// ===================================================================
// LayerNorm over last dim (256), one block per row
// ===================================================================
__global__ void k_layernorm(const float* __restrict__ pre, const float* __restrict__ gamma,
                            const float* __restrict__ beta, float* __restrict__ out)
{
  __shared__ float red[NOUT_];
  int row = blockIdx.x, t = threadIdx.x;
  float v = pre[(size_t)row * NOUT_ + t];
  red[t] = v; __syncthreads();
  for (int s = NOUT_ / 2; s > 0; s >>= 1) { if (t < s) red[t] += red[t + s]; __syncthreads(); }
  float mu = red[0] * (1.0f / NOUT_);
  __syncthreads();
  float d = v - mu;
  red[t] = d * d; __syncthreads();
  for (int s = NOUT_ / 2; s > 0; s >>= 1) { if (t < s) red[t] += red[t + s]; __syncthreads(); }
  float var = red[0] * (1.0f / NOUT_);
  out[(size_t)row * NOUT_ + t] = d / sqrtf(var + 1e-5f) * gamma[t] + beta[t];
}

// ===================================================================
extern "C" void kernel_launch(void* const* d_in, const int* in_sizes, int n_in,
                              void* d_out, int out_size, void* d_ws, size_t ws_size,
                              hipStream_t stream)
{
  (void)n_in; (void)out_size; (void)ws_size;
  const float* x     = (const float*)d_in[0];
  const int*   ei    = (const int*)d_in[1];
  const int*   batch = (const int*)d_in[2];
  const float* Ws[3]  = {(const float*)d_in[3],  (const float*)d_in[7],  (const float*)d_in[11]};
  const float* Wd[3]  = {(const float*)d_in[4],  (const float*)d_in[8],  (const float*)d_in[12]};
  const float* att[3] = {(const float*)d_in[5],  (const float*)d_in[9],  (const float*)d_in[13]};
  const float* bb[3]  = {(const float*)d_in[6],  (const float*)d_in[10], (const float*)d_in[14]};
  const float* Wh1 = (const float*)d_in[15]; const float* bh1 = (const float*)d_in[16];
  const float* Wh2 = (const float*)d_in[17]; const float* bh2 = (const float*)d_in[18];
  const float* gamma = (const float*)d_in[19]; const float* beta = (const float*)d_in[20];

  int n_nodes = in_sizes[2];
  int n_edges = in_sizes[1] / 2;
  int fin     = in_sizes[0] / n_nodes;   // 32
  int etot    = n_edges + n_nodes;       // edges + self-loops

  // ---- carve workspace (256B aligned) ----
  char* p = (char*)d_ws;
  auto carve = [&](size_t bytes) -> char* {
    char* r = p; p += (bytes + 255) & ~(size_t)255; return r;
  };
  float*    h      = (float*)carve((size_t)n_nodes * HID_ * 4);  // layer activations / agg target
  float*    hs     = (float*)carve((size_t)n_nodes * HID_ * 4);
  float*    hd     = (float*)carve((size_t)n_nodes * HID_ * 4);
  float*    ebuf   = (float*)carve((size_t)etot * HEADS * 4);
  unsigned* maxenc = (unsigned*)carve((size_t)n_nodes * HEADS * 4);
  float*    denom  = (float*)carve((size_t)n_nodes * HEADS * 4);
  float*    sums   = (float*)carve((size_t)GRAPHS * HID_ * 4);
  float*    cnts   = (float*)carve((size_t)GRAPHS * 4);
  float*    gbuf   = (float*)carve((size_t)GRAPHS * HID_ * 4);
  float*    mlph   = (float*)carve((size_t)GRAPHS * NHID_ * 4);
  float*    pre    = (float*)carve((size_t)GRAPHS * NOUT_ * 4);

  auto gemm = [&](const float* A, const float* W, float* out, int M, int N, int K,
                  const float* bias, int relu) {
    long long thr = (long long)(M / 16) * (N / 16) * 32;
    k_gemm16<<<(int)((thr + 255) / 256), 256, 0, stream>>>(A, W, out, M, N, K, bias, relu);
  };
  auto fill = [&](void* ptr, unsigned val, long long n) {
    k_fill_u32<<<(int)((n + 255) / 256), 256, 0, stream>>>((unsigned*)ptr, val, n);
  };

  long long ethr  = (long long)etot * 32;
  long long ehthr = (long long)etot * HEADS;
  long long nhthr = (long long)n_nodes * HID_;

  for (int layer = 0; layer < 3; ++layer) {
    const float* xin = (layer == 0) ? x : h;
    int K = (layer == 0) ? fin : HID_;

    gemm(xin, Ws[layer], hs, n_nodes, HID_, K, nullptr, 0);
    gemm(xin, Wd[layer], hd, n_nodes, HID_, K, nullptr, 0);

    fill(maxenc, ENC_NEG_MAX, (long long)n_nodes * HEADS);
    fill(denom, 0u, (long long)n_nodes * HEADS);

    k_edge_scores<<<(int)((ethr + 255) / 256), 256, 0, stream>>>(
        hs, hd, ei, n_edges, n_nodes, att[layer], ebuf, maxenc);

    fill(h, 0u, nhthr);   // GEMMs already consumed h; reuse as aggregation target

    k_edge_exp<<<(int)((ehthr + 255) / 256), 256, 0, stream>>>(
        ebuf, maxenc, ei, n_edges, n_nodes, denom);

    k_edge_aggr<<<(int)((ethr + 255) / 256), 256, 0, stream>>>(
        hs, ebuf, denom, ei, n_edges, n_nodes, h);

    k_bias_act<<<(int)((nhthr + 255) / 256), 256, 0, stream>>>(
        h, bb[layer], n_nodes, layer < 2 ? 1 : 0);
  }

  // global mean pool
  fill(sums, 0u, (long long)GRAPHS * HID_);
  fill(cnts, 0u, GRAPHS);
  k_pool<<<(int)((nhthr + 255) / 256), 256, 0, stream>>>(h, batch, sums, cnts, n_nodes);
  k_pool_div<<<(GRAPHS * HID_ + 255) / 256, 256, 0, stream>>>(sums, cnts, gbuf);

  // MLP head (WMMA) + LayerNorm
  gemm(gbuf, Wh1, mlph, GRAPHS, NHID_, HID_, bh1, 1);
  gemm(mlph, Wh2, pre,  GRAPHS, NOUT_, NHID_, bh2, 0);
  k_layernorm<<<GRAPHS, NOUT_, 0, stream>>>(pre, gamma, beta, (float*)d_out);
}